// FlexibleCategoricalSampler_87376814670201
// MI455X (gfx1250) — compile-verified
//
#include <hip/hip_runtime.h>
#include <stdint.h>

// Segmented Gumbel-argmax for MI455X (gfx1250, wave32).
// Pass A: per-group argmax via packed-u64 LDS ds_max atomics, merged with
//         512 global_atomic_max_u64 per block.
// Pass B: count items preceding each group's winner (== within-group rank).
// index[] is staged through LDS with global_load_async_to_lds_b128
// double-buffering (ASYNCcnt). logits/z use non-temporal b128 streaming loads
// so the 32MB index stream stays resident in the 192MB L2 for pass B.

#define NGROUPS 512
#define BLOCK   256
#define GRID    2048

typedef float vfloat4 __attribute__((ext_vector_type(4)));  // clang vector: legal
                                                            // for nontemporal builtin

__device__ __forceinline__ uint32_t ord_f32(float f) {
  // order-preserving map: float -> uint32 (handles negatives)
  uint32_t b = __float_as_uint(f);
  return (b & 0x80000000u) ? ~b : (b | 0x80000000u);
}

__device__ __forceinline__ uint32_t lds_lo32(const void* p) {
  // generic pointer to __shared__ has the wave-relative LDS offset in [31:0]
  return (uint32_t)(uintptr_t)p;
}

// CDNA5 async DMA global->LDS, tracked with ASYNCcnt. Default (RT) hint:
// keep the line in L2 for reuse by the next kernel.
__device__ __forceinline__ void async_ld_b128_rt(uint32_t lds_addr, const void* gaddr) {
  asm volatile("global_load_async_to_lds_b128 %0, %1, off"
               :
               : "v"(lds_addr), "v"((unsigned long long)(uintptr_t)gaddr)
               : "memory");
}

// Non-temporal flavor: last use of this data, don't retain in caches.
__device__ __forceinline__ void async_ld_b128_nt(uint32_t lds_addr, const void* gaddr) {
  asm volatile("global_load_async_to_lds_b128 %0, %1, off th:TH_LOAD_NT"
               :
               : "v"(lds_addr), "v"((unsigned long long)(uintptr_t)gaddr)
               : "memory");
}

__device__ __forceinline__ void wait_async_le1() {
  asm volatile("s_wait_asynccnt 1" ::: "memory");
}
__device__ __forceinline__ void wait_async_le0() {
  asm volatile("s_wait_asynccnt 0" ::: "memory");
}

__global__ void __launch_bounds__(BLOCK) gumbel_init_kernel(
    unsigned long long* __restrict__ gkeys, int* __restrict__ out) {
  int i = blockIdx.x * blockDim.x + threadIdx.x;
  if (i < NGROUPS) {
    gkeys[i] = 0ull;  // below any real packed key
    out[i]   = 0;
  }
}

// -------- Pass A: per-group (max y, min idx) as packed u64 --------
__global__ void __launch_bounds__(BLOCK) gumbel_argmax_kernel(
    const float* __restrict__ logits, const float* __restrict__ z,
    const int* __restrict__ index, unsigned long long* __restrict__ gkeys,
    int nvec /* number of float4 vectors = n/4 */) {
  __shared__ unsigned long long lkeys[NGROUPS];
  __shared__ __align__(16) int stage[2][BLOCK * 4];

  const int t = threadIdx.x;
  for (int g = t; g < NGROUPS; g += BLOCK) lkeys[g] = 0ull;
  __syncthreads();

  const int nchunks = nvec / BLOCK;  // chunk = BLOCK float4s
  const uint32_t st0 = lds_lo32(&stage[0][t * 4]);
  const uint32_t st1 = lds_lo32(&stage[1][t * 4]);

  int c = blockIdx.x;
  int buf = 0;
  if (c < nchunks) async_ld_b128_rt(st0, index + (size_t)(c * BLOCK + t) * 4);

  for (; c < nchunks; c += GRID) {
    const int cn = c + GRID;
    if (cn < nchunks) {
      async_ld_b128_rt(buf ? st0 : st1, index + (size_t)(cn * BLOCK + t) * 4);
      wait_async_le1();  // current buffer's DMA has landed in LDS
    } else {
      wait_async_le0();
    }

    const size_t v = (size_t)c * BLOCK + t;
    // Single-use streams: non-temporal so they don't evict index[] from L2.
    const vfloat4 lg = __builtin_nontemporal_load(&((const vfloat4*)logits)[v]);
    const vfloat4 zz = __builtin_nontemporal_load(&((const vfloat4*)z)[v]);
    const int4 gi = *(const int4*)&stage[buf][t * 4];
    const uint32_t base = (uint32_t)(v * 4);

    {
      unsigned long long k =
          ((unsigned long long)ord_f32(lg.x + zz.x) << 32) | (uint32_t)~(base + 0u);
      atomicMax(&lkeys[gi.x], k);
    }
    {
      unsigned long long k =
          ((unsigned long long)ord_f32(lg.y + zz.y) << 32) | (uint32_t)~(base + 1u);
      atomicMax(&lkeys[gi.y], k);
    }
    {
      unsigned long long k =
          ((unsigned long long)ord_f32(lg.z + zz.z) << 32) | (uint32_t)~(base + 2u);
      atomicMax(&lkeys[gi.z], k);
    }
    {
      unsigned long long k =
          ((unsigned long long)ord_f32(lg.w + zz.w) << 32) | (uint32_t)~(base + 3u);
      atomicMax(&lkeys[gi.w], k);
    }
    buf ^= 1;
  }

  __syncthreads();
  for (int g = t; g < NGROUPS; g += BLOCK) {
    const unsigned long long k = lkeys[g];
    if (k) atomicMax(&gkeys[g], k);  // global_atomic_max_u64, 512 slots
  }
}

// -------- Pass B: rank of winner = #items before it in same group --------
__global__ void __launch_bounds__(BLOCK) gumbel_rank_kernel(
    const int* __restrict__ index, const unsigned long long* __restrict__ gkeys,
    int* __restrict__ out, int nvec) {
  __shared__ int lcount[NGROUPS];
  __shared__ uint32_t widx[NGROUPS];
  __shared__ __align__(16) int stage[2][BLOCK * 4];

  const int t = threadIdx.x;
  for (int g = t; g < NGROUPS; g += BLOCK) {
    lcount[g] = 0;
    widx[g] = ~(uint32_t)(gkeys[g] & 0xFFFFFFFFull);  // winner's global index
  }
  __syncthreads();

  const int nchunks = nvec / BLOCK;
  const uint32_t st0 = lds_lo32(&stage[0][t * 4]);
  const uint32_t st1 = lds_lo32(&stage[1][t * 4]);

  int c = blockIdx.x;
  int buf = 0;
  if (c < nchunks) async_ld_b128_nt(st0, index + (size_t)(c * BLOCK + t) * 4);

  for (; c < nchunks; c += GRID) {
    const int cn = c + GRID;
    if (cn < nchunks) {
      // last use of index[]: non-temporal
      async_ld_b128_nt(buf ? st0 : st1, index + (size_t)(cn * BLOCK + t) * 4);
      wait_async_le1();
    } else {
      wait_async_le0();
    }

    const int4 gi = *(const int4*)&stage[buf][t * 4];
    const uint32_t base = (uint32_t)(((size_t)c * BLOCK + t) * 4);

    if (base + 0u < widx[gi.x]) atomicAdd(&lcount[gi.x], 1);
    if (base + 1u < widx[gi.y]) atomicAdd(&lcount[gi.y], 1);
    if (base + 2u < widx[gi.z]) atomicAdd(&lcount[gi.z], 1);
    if (base + 3u < widx[gi.w]) atomicAdd(&lcount[gi.w], 1);
    buf ^= 1;
  }

  __syncthreads();
  for (int g = t; g < NGROUPS; g += BLOCK) {
    const int cgt = lcount[g];
    if (cgt) atomicAdd(&out[g], cgt);
  }
}

extern "C" void kernel_launch(void* const* d_in, const int* in_sizes, int n_in,
                              void* d_out, int out_size, void* d_ws, size_t ws_size,
                              hipStream_t stream) {
  const float* logits = (const float*)d_in[0];
  const float* z      = (const float*)d_in[1];
  const int*   index  = (const int*)d_in[2];
  const int n = in_sizes[0];        // 8,388,608 (divisible by 4*BLOCK)
  const int nvec = n / 4;

  unsigned long long* gkeys = (unsigned long long*)d_ws;  // 512 * 8B
  int* out = (int*)d_out;                                 // 512 int32 ranks

  gumbel_init_kernel<<<(NGROUPS + BLOCK - 1) / BLOCK, BLOCK, 0, stream>>>(gkeys, out);
  gumbel_argmax_kernel<<<GRID, BLOCK, 0, stream>>>(logits, z, index, gkeys, nvec);
  gumbel_rank_kernel<<<GRID, BLOCK, 0, stream>>>(index, gkeys, out, nvec);
}